// Loss_12687333393051
// MI455X (gfx1250) — compile-verified
//
#include <hip/hip_runtime.h>
#include <hip/hip_bf16.h>
#include <math.h>

#define B_    32
#define N_    1514
#define M_    1602
#define C_    6
#define NOUT_ 512
#define EPS_  1e-7f
#define EMIX_ 0.1f

typedef __attribute__((ext_vector_type(2))) float v2f;
typedef __attribute__((ext_vector_type(8))) float v8f;
typedef __attribute__((ext_vector_type(4))) unsigned int u32x4;
typedef __attribute__((ext_vector_type(4))) int i32x4;
typedef __attribute__((ext_vector_type(8))) int i32x8;

#define WAVES   8
#define ITILE   16
#define JTILE   64
#define FSTRIDE 68                             // 64 data + 4 pad dwords per row
#define NTILES  ((M_ + JTILE - 1) / JTILE)     // 26
#define WROWS   1668                           // padded j extent (1668 % 64 == 4)

// ---------------------------------------------------------------------------
// TDM: DMA one 16-row x 64-col f32 tile (row stride M_) into LDS with a
// 68-float padded row stride; rows/cols beyond (rem_rows, rem_cols) zero-fill.
// D# packing per CDNA5 ISA.
// ---------------------------------------------------------------------------
static __device__ __forceinline__ void tdm_load_tile(
    unsigned lds_off, const float* gsrc, int rem_rows, int rem_cols)
{
  unsigned long long ga = (unsigned long long)(uintptr_t)gsrc;

  u32x4 g0;
  g0[0] = 1u;                                          // count=1, user descriptor
  g0[1] = lds_off;                                     // lds_addr (bytes)
  g0[2] = (unsigned)(ga & 0xFFFFFFFFull);              // global_addr[31:0]
  g0[3] = (unsigned)((ga >> 32) & 0x1FFFFFFull)        // global_addr[56:32]
        | (2u << 30);                                  // type = 2 ("image")

  unsigned d0 = (unsigned)rem_cols;                    // tensor_dim0 (remaining)
  unsigned d1 = (unsigned)rem_rows;                    // tensor_dim1 (remaining)
  i32x8 g1;
  g1[0] = (int)((2u << 16)      // data_size = 4 bytes
              | (1u << 20)      // pad_enable
              | (5u << 22)      // pad_interval: 2^(5+1) = 64 dwords
              | (3u << 25));    // pad_amount: 3+1 = 4 dwords -> 68-float stride
  g1[1] = (int)((d0 & 0xFFFFu) << 16);                 // tensor_dim0[15:0]
  g1[2] = (int)((d0 >> 16) | ((d1 & 0xFFFFu) << 16));  // dim0[31:16] | dim1[15:0]
  g1[3] = (int)((d1 >> 16) | ((unsigned)JTILE << 16)); // dim1[31:16] | tile_dim0
  g1[4] = (int)ITILE;                                  // tile_dim1=16, tile_dim2=0
  g1[5] = (int)M_;                                     // tensor_dim0_stride = 1602
  g1[6] = 0;                                           // stride0 hi | dim1_stride lo
  g1[7] = 0;                                           // dim1_stride hi

  i32x4 g2; g2[0] = 1; g2[1] = 1; g2[2] = 0; g2[3] = 0;          // dim2=1, dim3=1
  i32x4 g3; g3[0] = 0; g3[1] = (int)(1u << 16); g3[2] = 0; g3[3] = 0; // dim4=1

#if __has_include(<hip/amd_detail/amd_gfx1250_TDM.h>)
  i32x8 g4 = {0, 0, 0, 0, 0, 0, 0, 0};
  __builtin_amdgcn_tensor_load_to_lds(g0, g1, g2, g3, g4, 0);    // clang-23 lane
#else
  __builtin_amdgcn_tensor_load_to_lds(g0, g1, g2, g3, 0);        // ROCm 7.2 lane
#endif
}

// ---------------------------------------------------------------------------
// Kernel 1: loss2 via V_WMMA_F32_16X16X4_F32.
//   Accumulator cols 0..5 = f_pos = feature @ clip(w,0,1)
//                 cols 6..11 = f_neg = feature @ clip(w,-1,0), cols 12..15 = 0.
// Feature tiles arrive by double-buffered TDM (TENSORcnt); clipped weights
// live transposed in LDS (wcT[c'][j], col 12 = zeros) -> no in-loop barriers.
// ---------------------------------------------------------------------------
__global__ __launch_bounds__(256) void loss2_wmma_tdm_kernel(
    const float* __restrict__ y_pred_all,   // [B,N,C]
    const float* __restrict__ weight,       // [B,M,C]
    const float* __restrict__ feature,      // [B,N,M]
    float* __restrict__ loss2_accum)        // [1]
{
  __shared__ __align__(16) float ft[WAVES][2][ITILE * FSTRIDE];  // 69,632 B
  __shared__ float wcT[13][WROWS];                               // 86,736 B

  const int b      = blockIdx.y;
  const int w      = threadIdx.x >> 5;
  const int lane   = threadIdx.x & 31;
  const int i_base = blockIdx.x * (WAVES * ITILE) + w * ITILE;

  const int arow = lane & 15;          // A row / D column
  const int half = lane >> 4;
  const int asel = half << 1;          // K offset (0 or 2) for this lane half
  const int cp   = lane & 15;          // c' column in B / D
  const int cpc  = (cp < 12) ? cp : 12;// col 12 of wcT is all zeros (broadcast)

  // --- one-time: clipped weights, transposed, zero-padded ---
  for (int c = 0; c < 13; ++c) {
    for (int j = threadIdx.x; j < WROWS; j += 256) {
      float v = 0.f;
      if (c < 12 && j < M_) {
        float wv = weight[((size_t)b * M_ + j) * C_ + ((c < 6) ? c : (c - 6))];
        v = (c < 6) ? fminf(fmaxf(wv,  0.f), 1.f)
                    : fminf(fmaxf(wv, -1.f), 0.f);
      }
      wcT[c][j] = v;
    }
  }
  __syncthreads();   // only barrier in the kernel

  const int   rem_rows = (i_base < N_) ? (N_ - i_base) : 0;
  const int   ib       = (i_base < N_) ? i_base : 0;   // valid addr; rem_rows=0 -> no reads
  const float* gbase   = feature + ((size_t)b * N_ + ib) * M_;

  // prologue: tile 0 -> buffer 0
  tdm_load_tile((unsigned)(uintptr_t)(void*)&ft[w][0][0], gbase, rem_rows, M_);

  v8f acc = {0.f, 0.f, 0.f, 0.f, 0.f, 0.f, 0.f, 0.f};

  for (int t = 0; t < NTILES; ++t) {
    const int j0 = t * JTILE;

    if (t + 1 < NTILES) {
      tdm_load_tile((unsigned)(uintptr_t)(void*)&ft[w][(t + 1) & 1][0],
                    gbase + (j0 + JTILE), rem_rows, M_ - (j0 + JTILE));
      asm volatile("" ::: "memory");
      __builtin_amdgcn_s_wait_tensorcnt(1);   // tile t landed; t+1 in flight
      asm volatile("" ::: "memory");
    } else {
      asm volatile("" ::: "memory");
      __builtin_amdgcn_s_wait_tensorcnt(0);
      asm volatile("" ::: "memory");
    }

    const float* fbuf = &ft[w][t & 1][0];
    const float* wrow = &wcT[cpc][j0];

#pragma unroll
    for (int jj = 0; jj < JTILE; jj += 4) {
      float2 av = *(const float2*)&fbuf[arow * FSTRIDE + jj + asel];
      float2 bv = *(const float2*)&wrow[jj + asel];
      v2f A  = {av.x, av.y};
      v2f Bv = {bv.x, bv.y};
      acc = __builtin_amdgcn_wmma_f32_16x16x4_f32(
          /*neg_a=*/false, A, /*neg_b=*/false, Bv,
          /*c_mod=*/(short)0, acc, /*reuse_a=*/false, /*reuse_b=*/false);
    }
  }

  // --- epilogue: fold accumulator against y_pred_all, reduce, atomic add ---
  float part = 0.f;
#pragma unroll
  for (int r = 0; r < 8; ++r) {
    int   gi = i_base + r + 8 * half;   // D row = local i
    float av = acc[r];
    if (cp < 6) {
      float yp = (gi < N_) ? y_pred_all[((size_t)b * N_ + gi) * C_ + cp] : 0.f;
      part += (1.f - yp) * av;          // f_pos column
    } else if (cp < 12) {
      float yp = (gi < N_) ? y_pred_all[((size_t)b * N_ + gi) * C_ + (cp - 6)] : 0.f;
      part -= yp * av;                  // f_neg column
    }
  }
#pragma unroll
  for (int off = 16; off > 0; off >>= 1)
    part += __shfl_xor(part, off, 32);
  if (lane == 0)
    atomicAdd(loss2_accum, part);
}

// ---------------------------------------------------------------------------
// Kernel 2: batched gather + keras categorical CE + final mix with loss2.
// ---------------------------------------------------------------------------
__global__ __launch_bounds__(256) void loss1_kernel(
    const float* __restrict__ y_pred_all,   // [B,N,C]
    const float* __restrict__ y_true,       // [B,NOUT,C]
    const int*   __restrict__ out_indices,  // [B,NOUT]
    const float* __restrict__ loss2_ptr,    // [1]
    float* __restrict__ loss_out,           // [B,NOUT]
    float* __restrict__ y_pred_out)         // [B,NOUT,C]
{
  int t = blockIdx.x * blockDim.x + threadIdx.x;
  if (t >= B_ * NOUT_) return;
  int b   = t / NOUT_;
  int idx = out_indices[t];

  const float* row = y_pred_all + ((size_t)b * N_ + idx) * C_;
  float p[C_];
  float s = 0.f;
#pragma unroll
  for (int c = 0; c < C_; ++c) {
    p[c] = row[c];
    s += p[c];
    y_pred_out[(size_t)t * C_ + c] = p[c];
  }
  float inv = 1.f / s;
  float l1 = 0.f;
#pragma unroll
  for (int c = 0; c < C_; ++c) {
    float pc = p[c] * inv;
    pc = fminf(fmaxf(pc, EPS_), 1.f - EPS_);
    l1 -= y_true[(size_t)t * C_ + c] * logf(pc);
  }
  loss_out[t] = (1.f - EMIX_) * l1 + EMIX_ * loss2_ptr[0];
}

__global__ void zero_scalar_kernel(float* p) {
  if (threadIdx.x == 0 && blockIdx.x == 0) p[0] = 0.f;
}

// ---------------------------------------------------------------------------
extern "C" void kernel_launch(void* const* d_in, const int* in_sizes, int n_in,
                              void* d_out, int out_size, void* d_ws, size_t ws_size,
                              hipStream_t stream) {
  const float* y_pred_all  = (const float*)d_in[0];
  const float* y_true      = (const float*)d_in[1];
  const int*   out_indices = (const int*)  d_in[2];
  const float* weight      = (const float*)d_in[3];
  const float* feature     = (const float*)d_in[4];

  float* loss_out   = (float*)d_out;                 // [B,NOUT]
  float* y_pred_out = loss_out + (size_t)B_ * NOUT_; // [B,NOUT,C]
  float* loss2_acc  = (float*)d_ws;                  // 1 float scratch

  zero_scalar_kernel<<<1, 64, 0, stream>>>(loss2_acc);

  dim3 grid((N_ + WAVES * ITILE - 1) / (WAVES * ITILE), B_);
  loss2_wmma_tdm_kernel<<<grid, 256, 0, stream>>>(y_pred_all, weight, feature,
                                                  loss2_acc);

  int nt = B_ * NOUT_;
  loss1_kernel<<<(nt + 255) / 256, 256, 0, stream>>>(
      y_pred_all, y_true, out_indices, loss2_acc, loss_out, y_pred_out);
}